// OnnxAttDot_17489106829843
// MI455X (gfx1250) — compile-verified
//
#include <hip/hip_runtime.h>
#include <hip/hip_bf16.h>
#include <math.h>

// Problem constants (from reference)
#define DUNITS   1024
#define ATT_DIM  1024
#define EPROJS   1024
#define TLEN     32768
#define SCALING  2.0f

typedef __attribute__((ext_vector_type(2))) float v2f;
typedef __attribute__((ext_vector_type(8))) float v8f;

// ---------------------------------------------------------------------------
// K1: q = tanh(W_dec @ dec_z + b_dec) using V_WMMA_F32_16X16X4_F32.
// One wave computes 16 output rows. Per k-step of 4:
//   A slot (lane l, vgpr v) holds W[a0 + (l&15), k + v + 2*(l>>4)]   (ISA 16x4 layout)
//   B slot (lane l, vgpr v) holds dec_z[k + v + 2*(l>>4)] (replicated over N columns)
// D layout: vgpr j, lane 0  -> row a0+j ; lane 16 -> row a0+8+j.
// ---------------------------------------------------------------------------
__global__ void qproj_wmma_kernel(const float* __restrict__ W,
                                  const float* __restrict__ z,
                                  const float* __restrict__ bias,
                                  float* __restrict__ q) {
    const int lane = threadIdx.x & 31;
    const int wv   = threadIdx.x >> 5;                 // 4 waves / block
    const int a0   = (blockIdx.x * 4 + wv) * 16;       // 16 rows per wave
    const int m    = lane & 15;
    const int half = lane >> 4;

    const float* wrow = W + (size_t)(a0 + m) * DUNITS + 2 * half;
    const float* zp   = z + 2 * half;

    v8f acc = {};
    for (int k = 0; k < DUNITS; k += 4) {
        v2f a;
        a.x = wrow[k];
        a.y = wrow[k + 1];
        v2f b;
        b.x = zp[k];
        b.y = zp[k + 1];
        acc = __builtin_amdgcn_wmma_f32_16x16x4_f32(
            /*neg_a=*/false, a, /*neg_b=*/false, b,
            /*c_mod=*/(short)0, acc, /*reuse_a=*/false, /*reuse_b=*/false);
    }

    if (lane == 0) {
#pragma unroll
        for (int j = 0; j < 8; ++j)
            q[a0 + j] = tanhf(acc[j] + bias[a0 + j]);
    } else if (lane == 16) {
#pragma unroll
        for (int j = 0; j < 8; ++j)
            q[a0 + 8 + j] = tanhf(acc[j] + bias[a0 + 8 + j]);
    }
}

// ---------------------------------------------------------------------------
// K2: e[t] = dot(P[t,:], q) + mask[t].  One wave per row; float4 coalesced
// loads (512B contiguous per wave per iter), wave32 shuffle reduction.
// ---------------------------------------------------------------------------
__global__ void gemv_e_kernel(const float* __restrict__ P,
                              const float* __restrict__ q,
                              const float* __restrict__ mask,
                              float* __restrict__ e) {
    const int lane = threadIdx.x & 31;
    const int wv   = threadIdx.x >> 5;         // 8 waves / block
    const int row  = blockIdx.x * 8 + wv;

    const float4* prow = (const float4*)(P + (size_t)row * ATT_DIM);
    const float4* q4   = (const float4*)q;

    float acc = 0.0f;
#pragma unroll
    for (int it = 0; it < ATT_DIM / (4 * 32); ++it) {   // 8 iters
        const int idx = it * 32 + lane;
        const float4 a = prow[idx];
        const float4 b = q4[idx];
        acc += a.x * b.x + a.y * b.y + a.z * b.z + a.w * b.w;
    }
#pragma unroll
    for (int off = 16; off > 0; off >>= 1)
        acc += __shfl_xor(acc, off, 32);

    if (lane == 0)
        e[row] = acc + mask[row];
}

// ---------------------------------------------------------------------------
// K3: w = softmax(SCALING * e) over T.  Single block; e (128KB) is L2-hot.
// ---------------------------------------------------------------------------
__global__ void softmax_kernel(const float* __restrict__ e,
                               float* __restrict__ w) {
    __shared__ float sred[32];
    __shared__ float sM, sS;
    const int tid  = threadIdx.x;              // 1024 threads
    const int lane = tid & 31;
    const int wv   = tid >> 5;

    // pass 1: max
    float m = -1e30f;
    for (int i = tid; i < TLEN; i += 1024)
        m = fmaxf(m, SCALING * e[i]);
#pragma unroll
    for (int off = 16; off > 0; off >>= 1)
        m = fmaxf(m, __shfl_xor(m, off, 32));
    if (lane == 0) sred[wv] = m;
    __syncthreads();
    if (wv == 0) {
        float mm = sred[lane];
#pragma unroll
        for (int off = 16; off > 0; off >>= 1)
            mm = fmaxf(mm, __shfl_xor(mm, off, 32));
        if (lane == 0) sM = mm;
    }
    __syncthreads();
    const float M = sM;

    // pass 2: sum of exp
    float s = 0.0f;
    for (int i = tid; i < TLEN; i += 1024)
        s += expf(SCALING * e[i] - M);
#pragma unroll
    for (int off = 16; off > 0; off >>= 1)
        s += __shfl_xor(s, off, 32);
    if (lane == 0) sred[wv] = s;
    __syncthreads();
    if (wv == 0) {
        float ss = sred[lane];
#pragma unroll
        for (int off = 16; off > 0; off >>= 1)
            ss += __shfl_xor(ss, off, 32);
        if (lane == 0) sS = ss;
    }
    __syncthreads();
    const float inv = 1.0f / sS;

    // pass 3: write normalized weights
    for (int i = tid; i < TLEN; i += 1024)
        w[i] = expf(SCALING * e[i] - M) * inv;
}

// ---------------------------------------------------------------------------
// K4: partial c. 256 blocks x 128 rows each; each of 256 threads owns 4
// columns in registers; fully coalesced float4 stream over enc_h.
// Deterministic (no float atomics).
// ---------------------------------------------------------------------------
#define CBLOCKS 256
#define CROWS   (TLEN / CBLOCKS)   // 128

__global__ void gemv_c_partial_kernel(const float* __restrict__ enc,
                                      const float* __restrict__ w,
                                      float* __restrict__ part) {
    const int tid = threadIdx.x;               // 256 threads -> 1024 cols
    const int b   = blockIdx.x;
    const int t0  = b * CROWS;

    const float4* base = (const float4*)enc;
    float4 acc = make_float4(0.f, 0.f, 0.f, 0.f);

    for (int r = 0; r < CROWS; ++r) {
        const int t = t0 + r;
        const float wt = w[t];
        const float4 v = base[(size_t)t * (EPROJS / 4) + tid];
        acc.x += wt * v.x;
        acc.y += wt * v.y;
        acc.z += wt * v.z;
        acc.w += wt * v.w;
    }
    ((float4*)part)[(size_t)b * (EPROJS / 4) + tid] = acc;
}

// K5: c[j] = sum over blocks of partials (1MB read, negligible).
__global__ void reduce_c_kernel(const float* __restrict__ part,
                                float* __restrict__ c) {
    const int j = blockIdx.x * 256 + threadIdx.x;   // 4 blocks x 256 = 1024
    float s = 0.0f;
    for (int b = 0; b < CBLOCKS; ++b)
        s += part[(size_t)b * EPROJS + j];
    c[j] = s;
}

// ---------------------------------------------------------------------------
extern "C" void kernel_launch(void* const* d_in, const int* in_sizes, int n_in,
                              void* d_out, int out_size, void* d_ws, size_t ws_size,
                              hipStream_t stream) {
    const float* dec_z = (const float*)d_in[0];
    // d_in[1] = att_prev (unused by reference)
    const float* P     = (const float*)d_in[2];   // [T, ATT_DIM]
    const float* enc   = (const float*)d_in[3];   // [T, EPROJS]
    const float* mask  = (const float*)d_in[4];   // [T]
    const float* W     = (const float*)d_in[5];   // [ATT_DIM, DUNITS]
    const float* bias  = (const float*)d_in[6];   // [ATT_DIM]

    float* out_c = (float*)d_out;                 // [EPROJS]
    float* out_w = (float*)d_out + EPROJS;        // [T]

    float* ws   = (float*)d_ws;
    float* q    = ws;                             // 1024 floats
    float* e    = ws + ATT_DIM;                   // 32768 floats
    float* part = ws + ATT_DIM + TLEN;            // 256*1024 floats

    // K1: q projection via WMMA (64 waves total)
    qproj_wmma_kernel<<<ATT_DIM / (16 * 4), 128, 0, stream>>>(W, dec_z, bias, q);
    // K2: attention energies (128 MB stream)
    gemv_e_kernel<<<TLEN / 8, 256, 0, stream>>>(P, q, mask, e);
    // K3: softmax -> out_w
    softmax_kernel<<<1, 1024, 0, stream>>>(e, out_w);
    // K4: context partials (128 MB stream)
    gemv_c_partial_kernel<<<CBLOCKS, 256, 0, stream>>>(enc, out_w, part);
    // K5: final reduce -> out_c
    reduce_c_kernel<<<EPROJS / 256, 256, 0, stream>>>(part, out_c);
}